// CRF_59528246723081
// MI455X (gfx1250) — compile-verified
//
#include <hip/hip_runtime.h>
#include <math.h>

#define B_TOT 32
#define T_LEN 256
#define K_DIM 128
#define SOS 1
#define EOS 2
#define ROWS 16      // batches per workgroup (WMMA M dimension)
#define STR 132      // padded LDS row stride in dwords (132 % 64 == 4 -> conflict-free-ish)

typedef float v2f __attribute__((ext_vector_type(2)));
typedef float v8f __attribute__((ext_vector_type(8)));

// WGP-scope prefetch (scope 0 = pull into all cache levels, incl. WGP$).
// __builtin_prefetch can only reach SE/SYS scope, which skips the WGP cache.
__device__ __forceinline__ void prefetch_wgp(const void* p) {
    asm volatile("global_prefetch_b8 %0, off" :: "v"(p));
}

// ---------------- workspace init ----------------
__global__ void __launch_bounds__(64) crf_init_ws(float* ws) {
    if (threadIdx.x < 2) ws[threadIdx.x] = 0.0f;   // ws[0]=Z, ws[1]=gold
}

// ---------------- gold (emission+transition gather) ----------------
__global__ void __launch_bounds__(256) crf_gold(const float* __restrict__ h,
                                                const float* __restrict__ tr,
                                                const int* __restrict__ y,
                                                const int* __restrict__ lengths,
                                                float* __restrict__ ws) {
    int idx = blockIdx.x * 256 + threadIdx.x;
    int b = idx / T_LEN;
    int t = idx % T_LEN;
    float g = 0.0f;
    if (b < B_TOT && t < lengths[b]) {
        int yt = y[b * T_LEN + t];
        int py = (t == 0) ? SOS : y[b * T_LEN + t - 1];
        g = h[(size_t)b * T_LEN * K_DIM + (size_t)t * K_DIM + yt] + tr[py * K_DIM + yt];
    }
    __shared__ float red[256];
    red[threadIdx.x] = g;
    __syncthreads();
    for (int s = 128; s > 0; s >>= 1) {
        if (threadIdx.x < s) red[threadIdx.x] += red[threadIdx.x + s];
        __syncthreads();
    }
    if (threadIdx.x == 0) atomicAdd(&ws[1], red[0]);
}

// ---------------- forward scan: 2 blocks x 256 threads, 16 batches/block ----
// Per step: alpha_new(16x128) = log( P(16x128) @ E(128x128) ) + c + h[:,t,:]
// P = exp(alpha - rowmax), E = exp(transition) held in per-wave registers.
__global__ void __launch_bounds__(256) crf_forward_scan(const float* __restrict__ h,
                                                        const float* __restrict__ tr,
                                                        const int* __restrict__ lengths,
                                                        float* __restrict__ ws) {
    __shared__ float alphaS[ROWS][STR];
    __shared__ float pS[ROWS][STR];
    __shared__ float pmaxS[ROWS][17];
    __shared__ float cS[ROWS];
    __shared__ int   lenS[ROWS];

    const int tid   = threadIdx.x;
    const int lane  = tid & 31;
    const int wave  = tid >> 5;               // 0..7, owns 16 output columns
    const int bbase = blockIdx.x * ROWS;

    const int jcol  = wave * 16 + (lane & 15);  // this lane's output column (B/C/D N index)
    const int khalf = (lane >> 4) * 2;          // K sub-offset for A/B operand layout
    const int arow  = lane & 15;                // A-matrix row (M index) this lane supplies
    const int vrow0 = (lane >> 4) * 8;          // first C/D row this lane holds

    // ---- preload E = exp(transition) column slab into B-operand registers ----
    v2f eB[32];
#pragma unroll
    for (int kt = 0; kt < 32; ++kt) {
        int k = kt * 4 + khalf;
        v2f e;
        e.x = __expf(tr[(k + 0) * K_DIM + jcol]);
        e.y = __expf(tr[(k + 1) * K_DIM + jcol]);
        eB[kt] = e;
    }

    // ---- init: alpha0[b][j] = h[b,0,j] + tr[SOS][j]; load lengths ----
    if (tid < ROWS) lenS[tid] = lengths[bbase + tid];
    {
        int row = tid & 15, seg = tid >> 4;
        const float* hb = h + (size_t)(bbase + row) * T_LEN * K_DIM;
#pragma unroll
        for (int q = 0; q < 8; ++q) {
            int col = seg * 8 + q;
            alphaS[row][col] = hb[col] + tr[SOS * K_DIM + col];
        }
    }
    __syncthreads();

    for (int t = 1; t < T_LEN; ++t) {
        // ---- phase 1: per-row max (16 segments of 8 cols each) ----
        int row = tid & 15, seg = tid >> 4;
        float m = -3.4e38f;
#pragma unroll
        for (int q = 0; q < 8; ++q) m = fmaxf(m, alphaS[row][seg * 8 + q]);
        pmaxS[row][seg] = m;
        __syncthreads();
        if (tid < ROWS) {
            float c = pmaxS[tid][0];
#pragma unroll
            for (int s = 1; s < 16; ++s) c = fmaxf(c, pmaxS[tid][s]);
            cS[tid] = c;
        }
        __syncthreads();
        // ---- phase 2: P = exp(alpha - c) ----
        {
            float c = cS[row];
#pragma unroll
            for (int q = 0; q < 8; ++q) {
                int col = seg * 8 + q;
                pS[row][col] = __expf(alphaS[row][col] - c);
            }
        }
        __syncthreads();

        // ---- phase 3: WMMA 16x16 tile = P(16x128) @ E(128x16 slab) ----
        const float* hcol = h + (size_t)bbase * T_LEN * K_DIM + (size_t)t * K_DIM + jcol;
        if (t + 1 < T_LEN) {
            // WGP-scope prefetch of ALL next-step emission rows this lane will
            // read in phase 4: lands in WGP$, not just GL2.
#pragma unroll
            for (int v = 0; v < 8; ++v)
                prefetch_wgp(hcol + (size_t)(vrow0 + v) * T_LEN * K_DIM + K_DIM);
        }

        v8f acc0 = {}, acc1 = {}, acc2 = {}, acc3 = {};
#pragma unroll
        for (int kt = 0; kt < 32; kt += 4) {
            v2f a0 = { pS[arow][(kt + 0) * 4 + khalf], pS[arow][(kt + 0) * 4 + khalf + 1] };
            v2f a1 = { pS[arow][(kt + 1) * 4 + khalf], pS[arow][(kt + 1) * 4 + khalf + 1] };
            v2f a2 = { pS[arow][(kt + 2) * 4 + khalf], pS[arow][(kt + 2) * 4 + khalf + 1] };
            v2f a3 = { pS[arow][(kt + 3) * 4 + khalf], pS[arow][(kt + 3) * 4 + khalf + 1] };
            acc0 = __builtin_amdgcn_wmma_f32_16x16x4_f32(false, a0, false, eB[kt + 0],
                                                         (short)0, acc0, false, false);
            acc1 = __builtin_amdgcn_wmma_f32_16x16x4_f32(false, a1, false, eB[kt + 1],
                                                         (short)0, acc1, false, false);
            acc2 = __builtin_amdgcn_wmma_f32_16x16x4_f32(false, a2, false, eB[kt + 2],
                                                         (short)0, acc2, false, false);
            acc3 = __builtin_amdgcn_wmma_f32_16x16x4_f32(false, a3, false, eB[kt + 3],
                                                         (short)0, acc3, false, false);
        }
        v8f acc = (acc0 + acc1) + (acc2 + acc3);

        // ---- phase 4: alpha_new = log(acc) + c + h[b,t,j], masked update ----
#pragma unroll
        for (int v = 0; v < 8; ++v) {
            int r = vrow0 + v;
            float nv = __logf(acc[v]) + cS[r] + hcol[(size_t)r * T_LEN * K_DIM];
            if (t < lenS[r]) alphaS[r][jcol] = nv;
        }
        __syncthreads();
    }

    // ---- Z contribution: sum of alpha[b][EOS] over this block's batches ----
    if (tid == 0) {
        float z = 0.0f;
#pragma unroll
        for (int r = 0; r < ROWS; ++r) z += alphaS[r][EOS];
        atomicAdd(&ws[0], z);
    }
}

// ---------------- finalize ----------------
__global__ void __launch_bounds__(32) crf_finalize(const float* __restrict__ ws,
                                                   float* __restrict__ out) {
    if (threadIdx.x == 0) out[0] = (ws[0] - ws[1]) / (float)B_TOT;
}

extern "C" void kernel_launch(void* const* d_in, const int* in_sizes, int n_in,
                              void* d_out, int out_size, void* d_ws, size_t ws_size,
                              hipStream_t stream) {
    const float* h       = (const float*)d_in[0];
    const float* tr      = (const float*)d_in[1];
    const int*   y       = (const int*)d_in[2];
    const int*   lengths = (const int*)d_in[3];
    // d_in[4] (mask) is recomputed from lengths on device.
    float* ws  = (float*)d_ws;
    float* out = (float*)d_out;

    crf_init_ws<<<1, 64, 0, stream>>>(ws);
    crf_gold<<<(B_TOT * T_LEN + 255) / 256, 256, 0, stream>>>(h, tr, y, lengths, ws);
    crf_forward_scan<<<B_TOT / ROWS, 256, 0, stream>>>(h, tr, lengths, ws);
    crf_finalize<<<1, 32, 0, stream>>>(ws, out);
}